// IconCoGAT_41850161332739
// MI455X (gfx1250) — compile-verified
//
#include <hip/hip_runtime.h>

#define N_NODES   20000
#define E_PER_NET 320000
#define N_NETS    3
#define FEAT      128   // 4 heads * 32

typedef float v2f __attribute__((ext_vector_type(2)));
typedef float v8f __attribute__((ext_vector_type(8)));

// ---------------------------------------------------------------------------
// inv[n_id[i]] = i   (global->local remap; identity for n_id = arange)
// ---------------------------------------------------------------------------
__global__ void IconCoGAT_build_inverse(const int* __restrict__ n_id,
                                        int* __restrict__ inv) {
  int i = blockIdx.x * blockDim.x + threadIdx.x;
  if (i < N_NODES) inv[n_id[i]] = i;
}

// ---------------------------------------------------------------------------
// out[i, f] = bias[f]   (init accumulator; bias is zeros in reference)
// ---------------------------------------------------------------------------
__global__ void IconCoGAT_init_out(const float* __restrict__ bias,
                                   float* __restrict__ out) {
  int i = blockIdx.x * blockDim.x + threadIdx.x;
  if (i < N_NODES * FEAT) out[i] = bias[i & (FEAT - 1)];
}

// ---------------------------------------------------------------------------
// xs = x[n_id] @ W using V_WMMA_F32_16X16X4_F32.
// One wave per 16x16 output tile; 8 waves/block cover the 8 N-tiles of a row.
// A-frag (ISA 7.12.2, 32-bit A 16x4):  lane L holds (M=L%16, K=2*(L/16)+v)
// B-frag (mirrored):                   lane L holds (K=2*(L/16)+v, N=L%16)
// C/D  (32-bit 16x16): vgpr r, lane L -> (M = r + 8*(L/16), N = L%16)
// ---------------------------------------------------------------------------
__global__ __launch_bounds__(256) void IconCoGAT_gemm_xs(
    const float* __restrict__ x, const float* __restrict__ W,
    const int* __restrict__ n_id, float* __restrict__ xs) {
  const int lane = threadIdx.x & 31;
  const int wave = threadIdx.x >> 5;         // N tile: 0..7
  const int tm   = blockIdx.x;               // M tile: 0..1249
  const int m16  = lane & 15;
  const int koff = (lane >> 4) << 1;         // 0 or 2

  const int localRow = tm * 16 + m16;
  const int gidRow   = n_id[localRow];       // gather: xg = x[n_id]
  const float* arow  = x + (size_t)gidRow * FEAT;
  const int colB     = wave * 16 + m16;

  v8f c = {0.f, 0.f, 0.f, 0.f, 0.f, 0.f, 0.f, 0.f};
#pragma unroll
  for (int kb = 0; kb < FEAT; kb += 4) {
    v2f a = *(const v2f*)(arow + kb + koff);           // (m, k), (m, k+1)
    v2f b;
    b.x = W[(size_t)(kb + koff) * FEAT + colB];        // (k,   n)
    b.y = W[(size_t)(kb + koff + 1) * FEAT + colB];    // (k+1, n)
    c = __builtin_amdgcn_wmma_f32_16x16x4_f32(
        /*neg_a=*/false, a, /*neg_b=*/false, b,
        /*c_mod=*/(short)0, c, /*reuse_a=*/false, /*reuse_b=*/false);
  }

  const int nOut  = wave * 16 + m16;
  const int mBase = tm * 16 + (lane >> 4) * 8;
#pragma unroll
  for (int r = 0; r < 8; ++r) {
    xs[(size_t)(mBase + r) * FEAT + nOut] = c[r];
  }
}

// ---------------------------------------------------------------------------
// Edge scatter: one wave per edge across all 3 modalities.
// Lane l handles float4 at feature 4l (head = l/8).
// out[inv[dst]] += nw[m]*attn[e,h] * xs[inv[src]] via L2-side f32 atomics.
// ---------------------------------------------------------------------------
__global__ __launch_bounds__(256) void IconCoGAT_scatter_edges(
    const int* __restrict__ ei0, const int* __restrict__ ei1,
    const int* __restrict__ ei2, const float* __restrict__ a0,
    const float* __restrict__ a1, const float* __restrict__ a2,
    const float* __restrict__ nw, const int* __restrict__ inv,
    const float* __restrict__ xs, float* __restrict__ out) {
  const int lane = threadIdx.x & 31;
  const int wg   = blockIdx.x * (blockDim.x >> 5) + (threadIdx.x >> 5);
  if (wg >= N_NETS * E_PER_NET) return;      // uniform per-wave

  const int m = wg / E_PER_NET;
  const int e = wg - m * E_PER_NET;
  const int*   ei = (m == 0) ? ei0 : (m == 1) ? ei1 : ei2;
  const float* at = (m == 0) ? a0 : (m == 1) ? a1 : a2;

  const int src = ei[e];                     // edge_index[0, e]
  const int dst = ei[E_PER_NET + e];         // edge_index[1, e]
  const float alpha = nw[m] * at[(size_t)e * 4 + (lane >> 3)];

  const int sl = inv[src];
  const int dl = inv[dst];

  const float4 v = *(const float4*)(xs + (size_t)sl * FEAT + lane * 4);
  float* p = out + (size_t)dl * FEAT + lane * 4;

  const float r0 = alpha * v.x;
  const float r1 = alpha * v.y;
  const float r2 = alpha * v.z;
  const float r3 = alpha * v.w;

  // Non-returning hardware f32 atomics (L2-side), 4 dwords off one addr pair.
  asm volatile(
      "global_atomic_add_f32 %0, %1, off\n\t"
      "global_atomic_add_f32 %0, %2, off offset:4\n\t"
      "global_atomic_add_f32 %0, %3, off offset:8\n\t"
      "global_atomic_add_f32 %0, %4, off offset:12"
      :
      : "v"(p), "v"(r0), "v"(r1), "v"(r2), "v"(r3)
      : "memory");
}

// ---------------------------------------------------------------------------
extern "C" void kernel_launch(void* const* d_in, const int* in_sizes, int n_in,
                              void* d_out, int out_size, void* d_ws,
                              size_t ws_size, hipStream_t stream) {
  (void)in_sizes; (void)n_in; (void)out_size; (void)ws_size;

  const float* x    = (const float*)d_in[0];
  const int*   ei0  = (const int*)d_in[1];
  const int*   ei1  = (const int*)d_in[2];
  const int*   ei2  = (const int*)d_in[3];
  const float* a0   = (const float*)d_in[4];
  const float* a1   = (const float*)d_in[5];
  const float* a2   = (const float*)d_in[6];
  const float* nw   = (const float*)d_in[7];
  const float* W    = (const float*)d_in[8];
  const float* bias = (const float*)d_in[9];
  const int*   n_id = (const int*)d_in[10];

  float* out = (float*)d_out;
  float* xs  = (float*)d_ws;                                 // 10.24 MB
  int*   inv = (int*)((char*)d_ws + (size_t)N_NODES * FEAT * sizeof(float));

  IconCoGAT_build_inverse<<<(N_NODES + 255) / 256, 256, 0, stream>>>(n_id, inv);
  IconCoGAT_init_out<<<(N_NODES * FEAT + 255) / 256, 256, 0, stream>>>(bias, out);
  IconCoGAT_gemm_xs<<<N_NODES / 16, 256, 0, stream>>>(x, W, n_id, xs);

  const int totalWaves = N_NETS * E_PER_NET;                 // 960000
  IconCoGAT_scatter_edges<<<(totalWaves + 7) / 8, 256, 0, stream>>>(
      ei0, ei1, ei2, a0, a1, a2, nw, inv, xs, out);
}